// ESN_11390253269642
// MI455X (gfx1250) — compile-verified
//
#include <hip/hip_runtime.h>

// out[m, r] = (1 - lr) * tanh( sum_k x[m,k] * w_in[r,k] ),  m in [0, B*T), r in [0,128)
// M = 1,048,576, N = 128, K = 8.  Store-bandwidth bound (~536 MB out @ 23.3 TB/s).
// fp32 WMMA: V_WMMA_F32_16X16X4_F32, two chained ops cover K=8 exactly in fp32.

typedef __attribute__((ext_vector_type(2))) float v2f;
typedef __attribute__((ext_vector_type(8))) float v8f;

#define ESN_TOKENS (256 * 4096)   // B*T = 1,048,576
#define ESN_R 128
#define ESN_I 8

// Branch-free tanh: 1 - 2/(exp(2x)+1).  exp(2x)->inf gives 1, ->0 gives -1 (robust).
__device__ __forceinline__ float fast_tanh(float v) {
    float e = __expf(2.0f * v);                    // v_exp_f32 (fast)
    return 1.0f - 2.0f * __builtin_amdgcn_rcpf(e + 1.0f);
}

__global__ __launch_bounds__(256) void esn_wmma_kernel(
    const float* __restrict__ x,     // [TOKENS, 8] row-major
    const float* __restrict__ w_in,  // [128, 8]   row-major
    const float* __restrict__ lr,    // [1]
    float* __restrict__ out)         // [TOKENS, 128]
{
    const int lane = threadIdx.x & 31;
    const int wave = threadIdx.x >> 5;
    const long mtile = (long)blockIdx.x * 8 + wave;   // 16-token tile index

    const int half  = lane >> 4;      // 0: lanes 0..15, 1: lanes 16..31
    const int lid   = lane & 15;      // M (for A) / N (for B,C,D) within tile
    const int kbase = half * 2;       // ISA A/B layout: low lanes K={0,1}, high K={2,3}

    // ---- A fragments: 16 tokens x 8 inputs = two 16x4 f32 fragments -------
    const long tok = mtile * 16 + lid;
    const float* xp = x + tok * ESN_I;
    v2f a0 = __builtin_nontemporal_load((const v2f*)(xp + kbase));      // K = kbase..kbase+1
    v2f a1 = __builtin_nontemporal_load((const v2f*)(xp + kbase + 4));  // K = kbase+4..kbase+5

    // ---- B fragments for all 8 N-tiles: B[k][n] = w_in[n_glob*8 + k] ------
    v2f b0[8], b1[8];
#pragma unroll
    for (int nt = 0; nt < 8; ++nt) {
        const float* wp = w_in + (nt * 16 + lid) * ESN_I;
        b0[nt] = *(const v2f*)(wp + kbase);
        b1[nt] = *(const v2f*)(wp + kbase + 4);
    }

    const float scale = 1.0f - lr[0];   // = 0.5 with reference inputs

    // C/D layout: VGPR v, lane -> row M = v + half*8, col N = lid
    float* outp = out + (mtile * 16 + half * 8) * (long)ESN_R + lid;

#pragma unroll
    for (int nt = 0; nt < 8; ++nt) {
        v8f c = {};
        // 8 args: (neg_a, A, neg_b, B, c_mod, C, reuse_a, reuse_b)
        c = __builtin_amdgcn_wmma_f32_16x16x4_f32(false, a0, false, b0[nt],
                                                  (short)0, c, false, false);
        c = __builtin_amdgcn_wmma_f32_16x16x4_f32(false, a1, false, b1[nt],
                                                  (short)0, c, false, false);
#pragma unroll
        for (int v = 0; v < 8; ++v) {
            float r = scale * fast_tanh(c[v]);
            // 536 MB streaming output: bypass cache retention with NT stores.
            __builtin_nontemporal_store(r, outp + (long)v * ESN_R + nt * 16);
        }
    }
}

extern "C" void kernel_launch(void* const* d_in, const int* in_sizes, int n_in,
                              void* d_out, int out_size, void* d_ws, size_t ws_size,
                              hipStream_t stream) {
    const float* x    = (const float*)d_in[0];   // [B,T,I] fp32
    const float* w_in = (const float*)d_in[1];   // [R,I]   fp32
    // d_in[2] = d (diagonal) — multiplies a zero state in the reference; unused.
    const float* lr   = (const float*)d_in[3];   // [1]     fp32
    float* out = (float*)d_out;                  // [B,T,R] fp32

    // 65,536 M-tiles of 16 tokens; 8 waves (tiles) per 256-thread block.
    dim3 grid(ESN_TOKENS / 16 / 8), block(256);
    esn_wmma_kernel<<<grid, block, 0, stream>>>(x, w_in, lr, out);
}